// LocationEmbedding_44882408243821
// MI455X (gfx1250) — compile-verified
//
#include <hip/hip_runtime.h>
#include <hip/hip_bf16.h>

typedef __attribute__((ext_vector_type(2))) float v2f;
typedef __attribute__((ext_vector_type(8))) float v8f;

#define TPB 256

// ---------------------------------------------------------------------------
// K0: deg = 1.0 (self-loop weight), agg = 0
// ---------------------------------------------------------------------------
__global__ __launch_bounds__(TPB) void k_init(float* __restrict__ deg,
                                              float* __restrict__ agg,
                                              int n_nodes, int total) {
  int i = blockIdx.x * blockDim.x + threadIdx.x;
  if (i < total)   agg[i] = 0.0f;
  if (i < n_nodes) deg[i] = 1.0f;
}

// ---------------------------------------------------------------------------
// K1: weighted in-degree over edges
// ---------------------------------------------------------------------------
__global__ __launch_bounds__(TPB) void k_degree(const int* __restrict__ col,
                                                const float* __restrict__ w,
                                                float* __restrict__ deg, int E) {
  int e = blockIdx.x * blockDim.x + threadIdx.x;
  if (e < E) unsafeAtomicAdd(&deg[col[e]], w[e]);
}

// ---------------------------------------------------------------------------
// K2: dinv = rsqrt(deg) in place
// ---------------------------------------------------------------------------
__global__ __launch_bounds__(TPB) void k_dinv(float* __restrict__ deg, int n) {
  int i = blockIdx.x * blockDim.x + threadIdx.x;
  if (i < n) {
    float d = deg[i];
    deg[i] = (d > 0.0f) ? rsqrtf(d) : 0.0f;
  }
}

// ---------------------------------------------------------------------------
// K3: x = node_feat @ W via V_WMMA_F32_16X16X4_F32 (fp32 matrix cores).
// Block = 256 threads = 8 waves. Wave w computes 16x16 tile at
// rows [16*blockIdx.x, +16), cols [16*w, +16). 6250 blocks * 16 = 100000 rows.
// EXEC is all-ones everywhere (no divergence), as WMMA requires.
// ---------------------------------------------------------------------------
__global__ __launch_bounds__(TPB) void k_gemm(const float* __restrict__ A,
                                              const float* __restrict__ W,
                                              float* __restrict__ X) {
  const int wave = threadIdx.x >> 5;         // 0..7 -> column tile
  const int lane = threadIdx.x & 31;
  const int half = lane >> 4;                // 0: K={0,1}, 1: K={2,3}
  const int l16  = lane & 15;
  const int row0 = blockIdx.x * 16;
  const int col0 = wave * 16;

  const float* __restrict__ Arow = A + (size_t)(row0 + l16) * 128;
  v8f acc = {};
#pragma unroll
  for (int k = 0; k < 128; k += 4) {
    const int ka = k + 2 * half;
    v2f a, b;
    // A 16x4 frag: VGPR0 = A[M][ka], VGPR1 = A[M][ka+1]
    a.x = Arow[ka];
    a.y = Arow[ka + 1];
    // B 4x16 frag: VGPR0 = W[ka][N], VGPR1 = W[ka+1][N]
    b.x = W[(size_t)ka * 128 + col0 + l16];
    b.y = W[(size_t)(ka + 1) * 128 + col0 + l16];
    acc = __builtin_amdgcn_wmma_f32_16x16x4_f32(
        /*neg_a=*/false, a, /*neg_b=*/false, b,
        /*c_mod=*/(short)0, acc, /*reuse_a=*/false, /*reuse_b=*/false);
  }
  // C/D layout: VGPR v -> M = v + 8*half, N = l16
  float* __restrict__ out = X + (size_t)(row0 + half * 8) * 128 + col0 + l16;
#pragma unroll
  for (int v = 0; v < 8; ++v) out[(size_t)v * 128] = acc[v];
}

// ---------------------------------------------------------------------------
// K4: edge scatter. One wave per edge: lanes gather x[row] as float4 and do
// 4 hardware float atomics each into agg[col] (agg is L2-resident, 51 MB).
// ---------------------------------------------------------------------------
__global__ __launch_bounds__(TPB) void k_scatter(const int* __restrict__ rows,
                                                 const int* __restrict__ cols,
                                                 const float* __restrict__ w,
                                                 const float* __restrict__ dinv,
                                                 const float* __restrict__ X,
                                                 float* __restrict__ agg, int E) {
  const int e = blockIdx.x * (TPB / 32) + (threadIdx.x >> 5);
  if (e >= E) return;
  const int lane = threadIdx.x & 31;
  const int r = rows[e];
  const int c = cols[e];
  const float nrm = dinv[r] * w[e] * dinv[c];
  const float4 v = ((const float4*)(X + (size_t)r * 128))[lane];
  float* __restrict__ ag = agg + (size_t)c * 128 + lane * 4;
  unsafeAtomicAdd(ag + 0, nrm * v.x);
  unsafeAtomicAdd(ag + 1, nrm * v.y);
  unsafeAtomicAdd(ag + 2, nrm * v.z);
  unsafeAtomicAdd(ag + 3, nrm * v.w);
}

// ---------------------------------------------------------------------------
// K5: road = relu(agg + dinv^2 * x + b), in place into agg.
// (self-loop term dinv[i]*1*dinv[i]*x[i] folded in without atomics)
// ---------------------------------------------------------------------------
__global__ __launch_bounds__(TPB) void k_relu(const float* __restrict__ X,
                                              float* __restrict__ agg,
                                              const float* __restrict__ dinv,
                                              const float* __restrict__ bias,
                                              int total) {
  int idx = blockIdx.x * blockDim.x + threadIdx.x;
  if (idx >= total) return;
  const int i = idx >> 7;
  const int h = idx & 127;
  const float di = dinv[i];
  float v = agg[idx] + di * di * X[idx] + bias[h];
  agg[idx] = v > 0.0f ? v : 0.0f;
}

// ---------------------------------------------------------------------------
// K6: masked trajectory gather: out[b,l,:] = (l < seq_len[b]) ? road[idx] : 0
// ---------------------------------------------------------------------------
__global__ __launch_bounds__(TPB) void k_gather(const int* __restrict__ traj,
                                                const int* __restrict__ seq_len,
                                                const float* __restrict__ road,
                                                float* __restrict__ out,
                                                int L, int total) {
  int idx = blockIdx.x * blockDim.x + threadIdx.x;
  if (idx >= total) return;
  const int h  = idx & 127;
  const int bl = idx >> 7;
  const int b  = bl / L;
  const int l  = bl - b * L;
  float v = 0.0f;
  if (l < seq_len[b]) {
    const int node = traj[bl];
    v = road[(size_t)node * 128 + h];
  }
  out[idx] = v;
}

// ---------------------------------------------------------------------------
// Launcher
// inputs: 0 traj_seqs [B,L,1] i; 1 seq_len [B] i; 2 node_feat [N,128] f32;
//         3 edge_index [2,E] i; 4 edge_feat [E] f32; 5 W [128,128] f32; 6 b [128] f32
// output: [B, L, 128] f32
// ---------------------------------------------------------------------------
extern "C" void kernel_launch(void* const* d_in, const int* in_sizes, int n_in,
                              void* d_out, int out_size, void* d_ws, size_t ws_size,
                              hipStream_t stream) {
  const int*   traj      = (const int*)d_in[0];
  const int*   seq_len   = (const int*)d_in[1];
  const float* node_feat = (const float*)d_in[2];
  const int*   edge_idx  = (const int*)d_in[3];
  const float* edge_feat = (const float*)d_in[4];
  const float* W         = (const float*)d_in[5];
  const float* bias      = (const float*)d_in[6];

  const int N = in_sizes[2] / 128;       // 100000
  const int E = in_sizes[3] / 2;         // 1600000
  const int BL = in_sizes[0];            // B*L = 32768
  const int B  = in_sizes[1];            // 64
  const int L  = BL / B;                 // 512

  const int* rows = edge_idx;            // edge_index[0]
  const int* cols = edge_idx + E;        // edge_index[1]

  // workspace layout
  char* ws = (char*)d_ws;
  float* deg = (float*)ws;                                   // N floats (becomes dinv)
  size_t off_x = ((size_t)N * 4 + 4095) & ~(size_t)4095;
  float* x   = (float*)(ws + off_x);                         // N*128 floats
  float* agg = (float*)(ws + off_x + (size_t)N * 128 * 4);   // N*128 floats

  const int totalNH = N * 128;

  k_init<<<(totalNH + TPB - 1) / TPB, TPB, 0, stream>>>(deg, agg, N, totalNH);
  k_degree<<<(E + TPB - 1) / TPB, TPB, 0, stream>>>(cols, edge_feat, deg, E);
  k_dinv<<<(N + TPB - 1) / TPB, TPB, 0, stream>>>(deg, N);
  k_gemm<<<N / 16, TPB, 0, stream>>>(node_feat, W, x);       // 6250 blocks
  k_scatter<<<(E + (TPB / 32) - 1) / (TPB / 32), TPB, 0, stream>>>(
      rows, cols, edge_feat, deg, x, agg, E);
  k_relu<<<(totalNH + TPB - 1) / TPB, TPB, 0, stream>>>(x, agg, deg, bias, totalNH);
  k_gather<<<(out_size + TPB - 1) / TPB, TPB, 0, stream>>>(
      traj, seq_len, agg, (float*)d_out, L, out_size);
}